// QuantumFilter_65481071405431
// MI455X (gfx1250) — compile-verified
//
#include <hip/hip_runtime.h>
#include <hip/hip_bf16.h>

typedef __attribute__((ext_vector_type(2))) float v2f;
typedef __attribute__((ext_vector_type(8))) float v8f;

// ---------------------------------------------------------------------------
// Kernel 1: build M_q = Re(U^dagger Z_q U) for the fixed (param-only) part of
// the circuit. 16 threads each evolve one column of U (constant-index unrolled
// -> registers), then 256 threads reduce the 4 quadratic-form matrices.
// ---------------------------------------------------------------------------

template<int BIT>
__device__ __forceinline__ void apply1q(float ur[16], float ui[16],
    float g00r, float g00i, float g01r, float g01i,
    float g10r, float g10i, float g11r, float g11i) {
#pragma unroll
  for (int i0 = 0; i0 < 16; ++i0) {
    if (i0 & BIT) continue;
    const int i1 = i0 | BIT;
    const float ar = ur[i0], ai = ui[i0], br = ur[i1], bi = ui[i1];
    ur[i0] = g00r * ar - g00i * ai + g01r * br - g01i * bi;
    ui[i0] = g00r * ai + g00i * ar + g01r * bi + g01i * br;
    ur[i1] = g10r * ar - g10i * ai + g11r * br - g11i * bi;
    ui[i1] = g10r * ai + g10i * ar + g11r * bi + g11i * br;
  }
}

template<int CBIT, int TBIT>
__device__ __forceinline__ void applyCnot(float ur[16], float ui[16]) {
#pragma unroll
  for (int m = 0; m < 16; ++m) {
    if ((m & CBIT) && !(m & TBIT)) {
      const int m2 = m | TBIT;
      float tr = ur[m]; ur[m] = ur[m2]; ur[m2] = tr;
      float ti = ui[m]; ui[m] = ui[m2]; ui[m2] = ti;
    }
  }
}

__global__ void __launch_bounds__(256) qf_prep(const float* __restrict__ p,
                                               float* __restrict__ M) {
  __shared__ float sUr[16][16];
  __shared__ float sUi[16][16];
  const int t = threadIdx.x;
  if (t < 16) {
    float ur[16], ui[16];
#pragma unroll
    for (int m = 0; m < 16; ++m) { ur[m] = (m == t) ? 1.f : 0.f; ui[m] = 0.f; }
    float c, s;
    // RX(p0) wire0 (bit 8):  [[c,-is],[-is,c]]
    __sincosf(p[0] * 0.5f, &s, &c);
    apply1q<8>(ur, ui, c, 0.f, 0.f, -s, 0.f, -s, c, 0.f);
    // RY(p1) wire1 (bit 4):  [[c,-s],[s,c]]
    __sincosf(p[1] * 0.5f, &s, &c);
    apply1q<4>(ur, ui, c, 0.f, -s, 0.f, s, 0.f, c, 0.f);
    // RZ(p2) wire2 (bit 2):  diag(e^{-it/2}, e^{it/2})
    __sincosf(p[2] * 0.5f, &s, &c);
    apply1q<2>(ur, ui, c, -s, 0.f, 0.f, 0.f, 0.f, c, s);
    // RX(p3) wire3 (bit 1)
    __sincosf(p[3] * 0.5f, &s, &c);
    apply1q<1>(ur, ui, c, 0.f, 0.f, -s, 0.f, -s, c, 0.f);
    // CNOT(ctrl=wire0 bit8, tgt=wire1 bit4); CNOT(ctrl=wire2 bit2, tgt=wire3 bit1)
    applyCnot<8, 4>(ur, ui);
    applyCnot<2, 1>(ur, ui);
    // RY(p4) wire2
    __sincosf(p[4] * 0.5f, &s, &c);
    apply1q<2>(ur, ui, c, 0.f, -s, 0.f, s, 0.f, c, 0.f);
    // RZ(p5) wire0
    __sincosf(p[5] * 0.5f, &s, &c);
    apply1q<8>(ur, ui, c, -s, 0.f, 0.f, 0.f, 0.f, c, s);
#pragma unroll
    for (int m = 0; m < 16; ++m) { sUr[m][t] = ur[m]; sUi[m][t] = ui[m]; }
  }
  __syncthreads();
  const int j = t >> 4, k = t & 15;
#pragma unroll
  for (int q = 0; q < 4; ++q) {
    float acc = 0.f;
#pragma unroll
    for (int m = 0; m < 16; ++m) {
      const float zs = ((m >> (3 - q)) & 1) ? -1.f : 1.f;
      acc += zs * (sUr[m][j] * sUr[m][k] + sUi[m][j] * sUi[m][k]);
    }
    M[q * 256 + j * 16 + k] = acc;  // M_q row-major
  }
}

// ---------------------------------------------------------------------------
// Kernel 2: one wave per 16-sample tile.
//   D_q = M_q (A, 16x16) x Enc^T (B, 16x16, samples along N), via 4 chained
//   V_WMMA_F32_16X16X4_F32; then out[n,q] = sum_m D_q[m,n] * enc[n,m].
// Lane layout: every operand element lane l touches belongs to sample l%16.
// ---------------------------------------------------------------------------
__global__ void __launch_bounds__(256) qf_main(const float* __restrict__ x,
                                               const float* __restrict__ M,
                                               float* __restrict__ out,
                                               int nTiles) {
  const int lane = threadIdx.x & 31;
  const int tile = blockIdx.x * (blockDim.x >> 5) + (threadIdx.x >> 5);
  if (tile >= nTiles) return;               // wave-uniform exit
  const int n = lane & 15;                  // sample-in-tile owned by lane
  const bool hi = lane >= 16;
  const int s = tile * 16 + n;              // global sample index

  // x is [B,4] row-major -> one coalesced float4 per lane
  const float4 xv = reinterpret_cast<const float4*>(x)[s];
  float c0, s0, c1, s1, c2, s2, c3, s3;
  __sincosf(xv.x * 0.5f, &s0, &c0);
  __sincosf(xv.y * 0.5f, &s1, &c1);
  __sincosf(xv.z * 0.5f, &s2, &c2);
  __sincosf(xv.w * 0.5f, &s3, &c3);

  // enc[k] = prod_i (bit_i(k) ? sin : cos), qubit 0 = MSB (bit 8)
  const float p00 = c0 * c1, p01 = c0 * s1, p10 = s0 * c1, p11 = s0 * s1;
  const float q00 = c2 * c3, q01 = c2 * s3, q10 = s2 * c3, q11 = s2 * s3;
  const float e0  = p00 * q00, e1  = p00 * q01, e2  = p00 * q10, e3  = p00 * q11;
  const float e4  = p01 * q00, e5  = p01 * q01, e6  = p01 * q10, e7  = p01 * q11;
  const float e8  = p10 * q00, e9  = p10 * q01, e10 = p10 * q10, e11 = p10 * q11;
  const float e12 = p11 * q00, e13 = p11 * q01, e14 = p11 * q10, e15 = p11 * q11;

  // B-operand K-chunks (4x16 slices): vgpr0 holds row 4c+(hi?2:0), vgpr1 row 4c+(hi?3:1)
  v2f b0 = { hi ? e2  : e0,  hi ? e3  : e1  };
  v2f b1 = { hi ? e6  : e4,  hi ? e7  : e5  };
  v2f b2 = { hi ? e10 : e8,  hi ? e11 : e9  };
  v2f b3 = { hi ? e14 : e12, hi ? e15 : e13 };

  // elementwise partner for rows m = r + 8*hi of the D (C-layout) tile
  const float em0 = hi ? e8  : e0, em1 = hi ? e9  : e1;
  const float em2 = hi ? e10 : e2, em3 = hi ? e11 : e3;
  const float em4 = hi ? e12 : e4, em5 = hi ? e13 : e5;
  const float em6 = hi ? e14 : e6, em7 = hi ? e15 : e7;

  const int row = n;  // A-matrix row held by this lane
  float o0 = 0.f, o1 = 0.f, o2 = 0.f, o3 = 0.f;
#pragma unroll
  for (int q = 0; q < 4; ++q) {
    const float4* mrow = reinterpret_cast<const float4*>(M + q * 256 + row * 16);
    v8f acc = {0.f, 0.f, 0.f, 0.f, 0.f, 0.f, 0.f, 0.f};
#pragma unroll
    for (int c = 0; c < 4; ++c) {
      const float4 m4 = mrow[c];                 // M_q[row][4c..4c+3], L2-resident
      v2f a = { hi ? m4.z : m4.x, hi ? m4.w : m4.y };
      v2f bb = (c == 0) ? b0 : (c == 1) ? b1 : (c == 2) ? b2 : b3;
      acc = __builtin_amdgcn_wmma_f32_16x16x4_f32(
          /*neg_a=*/false, a, /*neg_b=*/false, bb,
          /*c_mod=*/(short)0, acc, /*reuse_a=*/false, /*reuse_b=*/false);
    }
    float v = acc[0] * em0 + acc[1] * em1 + acc[2] * em2 + acc[3] * em3 +
              acc[4] * em4 + acc[5] * em5 + acc[6] * em6 + acc[7] * em7;
    v += __shfl_xor(v, 16, 32);                  // combine m=0..7 with m=8..15
    if (q == 0) o0 = v; else if (q == 1) o1 = v; else if (q == 2) o2 = v; else o3 = v;
  }

  if (!hi) {                                     // lanes 0..15: coalesced float4 store
    reinterpret_cast<float4*>(out)[s] = make_float4(o0, o1, o2, o3);
  }
}

// ---------------------------------------------------------------------------
extern "C" void kernel_launch(void* const* d_in, const int* in_sizes, int n_in,
                              void* d_out, int out_size, void* d_ws, size_t ws_size,
                              hipStream_t stream) {
  const float* x      = (const float*)d_in[0];   // [B,4] f32
  const float* params = (const float*)d_in[1];   // [6]   f32
  float* out = (float*)d_out;                    // [B,4] f32
  float* M   = (float*)d_ws;                     // 4*16*16 f32 scratch

  const int B = in_sizes[0] / 4;
  const int nTiles = B / 16;                     // B = 524288 -> 32768 tiles

  qf_prep<<<1, 256, 0, stream>>>(params, M);
  const int wavesPerBlock = 8;                   // 256 threads = 8 wave32
  const int blocks = (nTiles + wavesPerBlock - 1) / wavesPerBlock;
  qf_main<<<blocks, 256, 0, stream>>>(x, M, out, nTiles);
}